// CvxpyLayersSolver_52621939311031
// MI455X (gfx1250) — compile-verified
//
#include <hip/hip_runtime.h>

typedef __attribute__((ext_vector_type(2))) float v2f;
typedef __attribute__((ext_vector_type(8))) float v8f;

#define EPS 1e-8f

// D = A(16x4) x B(4x16) + C, f32 WMMA.
// A = G zero-padded to 16x4 (M = edge row, K = state dim)
// B = P^T               (K = state dim, N = point within tile)
// C = -h broadcast per row M
// Resulting D[e][n] = (G p_n - h)[e] with point n's 4 edge values all living in
// lane n (VGPRs 0..3), so relu / w = G^T v / norm / scale / stores are lane-local.
__global__ void __launch_bounds__(256)
cvxpy_box_dual_kernel(const float* __restrict__ pc,   // (N,2) row-major
                      const float* __restrict__ G,    // (4,3) row-major
                      const float* __restrict__ h,    // (4,)
                      float* __restrict__ out,        // mu.T (4,N) ++ lam.T (3,N)
                      int N) {
    const int lane = threadIdx.x & 31;
    const int wave = (blockIdx.x * blockDim.x + threadIdx.x) >> 5;
    const int m0   = wave << 4;              // 16 points per wave-tile
    if (m0 >= N) return;                     // wave-uniform: EXEC stays full

    const int l    = lane & 15;
    const bool lo  = (lane < 16);
    const int idx  = m0 + l;

    // ---- A = G padded (VGPR0: K=0 / K=2, VGPR1: K=1 / K=3) ----
    v2f a;
    if (lo) {
        a.x = (l < 4) ? G[l * 3 + 0] : 0.0f;     // K=0
        a.y = (l < 4) ? G[l * 3 + 1] : 0.0f;     // K=1
    } else {
        a.x = (l < 4) ? G[l * 3 + 2] : 0.0f;     // K=2
        a.y = 0.0f;                              // K=3 (pad)
    }

    // ---- B = P^T: lane l (lower half) holds point m0+l; upper half = aug/pad zeros ----
    v2f b; b.x = 0.0f; b.y = 0.0f;
    if (lo && idx < N) {
        const float2 p = ((const float2* __restrict__)pc)[idx];  // coalesced 8B/lane
        b.x = p.x;                               // K=0
        b.y = p.y;                               // K=1
    }

    // ---- C = -h per row M (M = j for lanes 0-15, j+8 for lanes 16-31) ----
    v8f c;
#pragma unroll
    for (int j = 0; j < 8; ++j) {
        const int M = j + (lo ? 0 : 8);
        c[j] = (M < 4) ? -h[M] : 0.0f;
    }

    // ---- r = G p - h via WMMA (full EXEC here) ----
    v8f d = __builtin_amdgcn_wmma_f32_16x16x4_f32(
        /*neg_a=*/false, a, /*neg_b=*/false, b,
        /*c_mod=*/(short)0, c, /*reuse_a=*/false, /*reuse_b=*/false);

    // ---- lane-local epilogue: only lower half holds the meaningful rows 0..3 ----
    if (lo && idx < N) {
        const float v0 = fmaxf(d[0], 0.0f);
        const float v1 = fmaxf(d[1], 0.0f);
        const float v2 = fmaxf(d[2], 0.0f);
        const float v3 = fmaxf(d[3], 0.0f);

        // w = G^T v  (S = 3)
        const float wx = v0 * G[0] + v1 * G[3] + v2 * G[6] + v3 * G[9];
        const float wy = v0 * G[1] + v1 * G[4] + v2 * G[7] + v3 * G[10];
        const float wz = v0 * G[2] + v1 * G[5] + v2 * G[8] + v3 * G[11];

        const float dist = sqrtf(wx * wx + wy * wy + wz * wz);
        const float inv  = (dist > EPS) ? (1.0f / dist) : 0.0f;

        // mu.T: (4, N) row-major -> out[e*N + idx]; stores coalesced over lanes 0-15
        out[0 * N + idx] = v0 * inv;
        out[1 * N + idx] = v1 * inv;
        out[2 * N + idx] = v2 * inv;
        out[3 * N + idx] = v3 * inv;

        // lam.T: (3, N) at offset 4N
        float* __restrict__ lam = out + (size_t)4 * (size_t)N;
        lam[0 * N + idx] = -wx * inv;
        lam[1 * N + idx] = -wy * inv;
        lam[2 * N + idx] = -wz * inv;
    }
}

extern "C" void kernel_launch(void* const* d_in, const int* in_sizes, int n_in,
                              void* d_out, int out_size, void* d_ws, size_t ws_size,
                              hipStream_t stream) {
    const float* pc = (const float*)d_in[0];  // (N,2) float32
    const float* G  = (const float*)d_in[1];  // (4,3) float32
    const float* h  = (const float*)d_in[2];  // (4,)  float32
    float* out      = (float*)d_out;          // 7*N float32

    const int N = in_sizes[0] / 2;
    const int tiles = (N + 15) / 16;          // 16 points per wave
    const int waves_per_block = 256 / 32;     // wave32: 8 waves / block
    const int blocks = (tiles + waves_per_block - 1) / waves_per_block;

    cvxpy_box_dual_kernel<<<blocks, 256, 0, stream>>>(pc, G, h, out, N);
}